// TiRGN_74698071212221
// MI455X (gfx1250) — compile-verified
//
#include <hip/hip_runtime.h>
#include <hip/hip_bf16.h>

#define N_ENT 20000
#define N_REL2 400
#define DIM 128
#define NB 8
#define NL 2
#define T_SNAP 8
#define NE 100000

typedef float v2f __attribute__((ext_vector_type(2)));
typedef float v8f __attribute__((ext_vector_type(8)));

// CDNA5 async global->LDS copy (ASYNCcnt-tracked). VDST = LDS byte address,
// VADDR = 64-bit global address. Generic pointers to __shared__ carry the LDS
// offset in their low 32 bits.
__device__ __forceinline__ void async_copy_b128(unsigned lds_addr, const float* gptr) {
    asm volatile("global_load_async_to_lds_b128 %0, %1, off"
                 :: "v"(lds_addr), "v"(gptr)
                 : "memory");
}
__device__ __forceinline__ void wait_async0() {
    asm volatile("s_wait_asynccnt 0x0" ::: "memory");
}

// ---------------------------------------------------------------------------
// Repack basis (L,B,D,D) -> Wb (L, K=D*B, D) with K index k = d*8+b, so that
// nbr[n,e] = sum_k agg[n,k] * Wb[l][k][e]  with agg laid out (N, D, B).
// ---------------------------------------------------------------------------
__global__ __launch_bounds__(256) void repack_basis_k(const float* __restrict__ basis,
                                                      float* __restrict__ Wb) {
    int i = blockIdx.x * 256 + threadIdx.x;            // over NL * (D*NB) * D
    if (i >= NL * DIM * NB * DIM) return;
    int e = i % DIM;
    int k = (i / DIM) % (DIM * NB);
    int l = i / (DIM * NB * DIM);
    int d = k >> 3, b = k & 7;
    Wb[((size_t)l * DIM * NB + k) * DIM + e] = basis[((((size_t)l * NB) + b) * DIM + d) * DIM + e];
}

// ---------------------------------------------------------------------------
// Relation context: masked scatter of source features per relation type.
// One wave (32 lanes) per edge, float4 per lane (32*4 = 128 = D).
// ---------------------------------------------------------------------------
__global__ __launch_bounds__(256) void rel_scatter_k(const float* __restrict__ h,
                                                     const int* __restrict__ src,
                                                     const int* __restrict__ et,
                                                     const float* __restrict__ em,
                                                     float* __restrict__ rel_sum,
                                                     float* __restrict__ rel_cnt) {
    int e = blockIdx.x * 8 + (threadIdx.x >> 5);
    if (e >= NE) return;
    int lane = threadIdx.x & 31;
    float m = em[e];
    if (m == 0.0f) return;
    int s = src[e];
    int r = et[e];
    float4 v = ((const float4*)(h + (size_t)s * DIM))[lane];
    float* out = rel_sum + (size_t)r * DIM + lane * 4;
    atomicAdd(out + 0, v.x * m);
    atomicAdd(out + 1, v.y * m);
    atomicAdd(out + 2, v.z * m);
    atomicAdd(out + 3, v.w * m);
    if (lane == 0) atomicAdd(rel_cnt + r, m);
}

// ---------------------------------------------------------------------------
// Per-relation projection + GRU (400 rows x 128 dim; tiny -> VALU kernel).
// ---------------------------------------------------------------------------
__global__ __launch_bounds__(128) void rel_gru_k(const float* __restrict__ rel_sum,
                                                 const float* __restrict__ rel_cnt,
                                                 const float* __restrict__ proj_w,
                                                 const float* __restrict__ proj_b,
                                                 const float* __restrict__ r_in,
                                                 const float* __restrict__ wi,
                                                 const float* __restrict__ wh,
                                                 const float* __restrict__ bb,
                                                 float* __restrict__ r_out) {
    __shared__ float mean[DIM];
    __shared__ float rp[DIM];
    __shared__ float hrow[DIM];
    int rel = blockIdx.x;
    int t = threadIdx.x;
    float cnt = rel_cnt[rel];
    cnt = cnt < 1.0f ? 1.0f : cnt;
    mean[t] = rel_sum[(size_t)rel * DIM + t] / cnt;
    hrow[t] = r_in[(size_t)rel * DIM + t];
    __syncthreads();
    float acc = proj_b[t];
    for (int k = 0; k < DIM; ++k) acc += mean[k] * proj_w[k * DIM + t];
    rp[t] = acc;
    __syncthreads();
    float xr = 0.f, xz = 0.f, xn = 0.f, hr = 0.f, hz = 0.f, hn = 0.f;
    for (int k = 0; k < DIM; ++k) {
        float xv = rp[k], hv = hrow[k];
        const float* wik = wi + (size_t)k * 3 * DIM;
        const float* whk = wh + (size_t)k * 3 * DIM;
        xr += xv * wik[t];          hr += hv * whk[t];
        xz += xv * wik[DIM + t];    hz += hv * whk[DIM + t];
        xn += xv * wik[2*DIM + t];  hn += hv * whk[2*DIM + t];
    }
    float rg = 1.0f / (1.0f + __expf(-(xr + hr + bb[t])));
    float zg = 1.0f / (1.0f + __expf(-(xz + hz + bb[DIM + t])));
    float ng = tanhf(xn + rg * hn + bb[2*DIM + t]);
    r_out[(size_t)rel * DIM + t] = (1.0f - zg) * ng + zg * hrow[t];
}

// ---------------------------------------------------------------------------
// R-GCN edge scatter: agg[dst, d, b] += comp[et,b] * x[src,d] * em.
// B innermost so each lane's 8 atomics hit 8 consecutive dwords (L2-resident).
// ---------------------------------------------------------------------------
__global__ __launch_bounds__(256) void edge_scatter_k(const float* __restrict__ x,
                                                      const int* __restrict__ src,
                                                      const int* __restrict__ dst,
                                                      const int* __restrict__ et,
                                                      const float* __restrict__ em,
                                                      const float* __restrict__ comp_l,
                                                      float* __restrict__ agg,
                                                      float* __restrict__ deg,
                                                      int addDeg) {
    int e = blockIdx.x * 2 + (threadIdx.x >> 7);
    if (e >= NE) return;
    int d = threadIdx.x & 127;
    float m = em[e];
    if (m == 0.0f) return;
    int s = src[e], dn = dst[e], r = et[e];
    float xs = x[(size_t)s * DIM + d] * m;
    const float* c = comp_l + (size_t)r * NB;
    float* a = agg + (size_t)dn * (DIM * NB) + d * NB;
#pragma unroll
    for (int b = 0; b < NB; ++b) atomicAdd(a + b, c[b] * xs);
    if (addDeg && d == 0) atomicAdd(deg + dn, m);
}

// ---------------------------------------------------------------------------
// Tiled f32 WMMA GEMM: C[M x Nc] = A[M x K] @ Bm[K x Nc], row-major.
// Block = 256 threads (8 waves). Each block: 8 row tiles (128 rows) x one
// 64-column group. B is staged through LDS in 64-K-row chunks via CDNA5
// async global->LDS copies, double-buffered (copy chunk c+1 overlaps WMMA on
// chunk c; s_wait_asynccnt + barrier between). Row stride 72 dwords makes all
// B-fragment ds_load_b32 patterns bank-conflict-free. A fragments for a chunk
// are batch-loaded into registers (16 independent float2 loads).
// Requires M%16==0, K%64==0, Nc%64==0.
// V_WMMA_F32_16X16X4_F32 fragments:
//   A 16x4:  lane m=lane%16, K = vgpr + 2*(lane/16)
//   B 4x16:  lane n=lane%16, K = vgpr + 2*(lane/16)
//   C 16x16: vgpr v -> M = v + 8*(lane/16), N = lane%16
// ---------------------------------------------------------------------------
#define KC 64
#define BSTRIDE 72

__global__ __launch_bounds__(256) void gemm_wmma_k(const float* __restrict__ A,
                                                   const float* __restrict__ Bm,
                                                   float* __restrict__ C,
                                                   int M, int K, int Nc) {
    __shared__ float Bs[2][KC * BSTRIDE];   // 2 x 18 KB

    int Mt = M >> 4, Ng = Nc >> 6;
    int mb = blockIdx.x / Ng;
    int ng = blockIdx.x % Ng;
    int tid = threadIdx.x;
    int wv = tid >> 5;
    int mt = mb * 8 + wv;
    bool active = (mt < Mt);
    int lane = tid & 31;
    int m16 = lane & 15, half = lane >> 4;

    int nchunks = K / KC;

    // issue async copy of B chunk `c` into buffer `d` (1024 b128 segments)
    auto copy_chunk = [&](int c, int d) {
        int seg0 = tid;                       // 4 segments per thread
#pragma unroll
        for (int i = 0; i < 4; ++i) {
            int seg = seg0 + i * 256;
            int row = seg >> 4;               // 0..63
            int col = (seg & 15) * 4;         // 0..60
            const float* gsrc = Bm + (size_t)(c * KC + row) * Nc + ng * 64 + col;
            unsigned ldst = (unsigned)(size_t)&Bs[d][row * BSTRIDE + col];
            async_copy_b128(ldst, gsrc);
        }
    };

    v8f c0 = {0.f,0.f,0.f,0.f,0.f,0.f,0.f,0.f};
    v8f c1 = c0, c2 = c0, c3 = c0;

    const float* arow = A + (size_t)(mt * 16 + m16) * K + 2 * half;

    copy_chunk(0, 0);
    wait_async0();
    __syncthreads();

    for (int c = 0; c < nchunks; ++c) {
        int cur = c & 1;
        if (c + 1 < nchunks) copy_chunk(c + 1, cur ^ 1);

        if (active) {
            // batch-load this chunk's A fragments into registers
            v2f areg[KC / 4];
            const float* ac = arow + c * KC;
#pragma unroll
            for (int i = 0; i < KC / 4; ++i) areg[i] = *(const v2f*)(ac + 4 * i);

            const float* bs = &Bs[cur][0];
#pragma unroll
            for (int i = 0; i < KC / 4; ++i) {
                const float* bp = bs + (4 * i + 2 * half) * BSTRIDE + m16;
                v2f b;
                b.x = bp[0];            b.y = bp[BSTRIDE];
                c0 = __builtin_amdgcn_wmma_f32_16x16x4_f32(false, areg[i], false, b, (short)0, c0, false, false);
                b.x = bp[16];           b.y = bp[BSTRIDE + 16];
                c1 = __builtin_amdgcn_wmma_f32_16x16x4_f32(false, areg[i], false, b, (short)0, c1, false, false);
                b.x = bp[32];           b.y = bp[BSTRIDE + 32];
                c2 = __builtin_amdgcn_wmma_f32_16x16x4_f32(false, areg[i], false, b, (short)0, c2, false, false);
                b.x = bp[48];           b.y = bp[BSTRIDE + 48];
                c3 = __builtin_amdgcn_wmma_f32_16x16x4_f32(false, areg[i], false, b, (short)0, c3, false, false);
            }
        }

        wait_async0();       // this wave's copies for chunk c+1 have landed
        __syncthreads();     // everyone done reading `cur` / copies visible
    }

    if (active) {
        int rbase = mt * 16 + 8 * half;
        int cbase = ng * 64 + m16;
#pragma unroll
        for (int v = 0; v < 8; ++v) {
            float* crow = C + (size_t)(rbase + v) * Nc + cbase;
            crow[0]  = c0[v];
            crow[16] = c1[v];
            crow[32] = c2[v];
            crow[48] = c3[v];
        }
    }
}

// ---------------------------------------------------------------------------
// x_out = relu(nbr/deg + selfp), in-place into nbr.
// ---------------------------------------------------------------------------
__global__ __launch_bounds__(256) void combine_layer_k(float* __restrict__ nbr,
                                                       const float* __restrict__ selfp,
                                                       const float* __restrict__ deg) {
    int i = blockIdx.x * 256 + threadIdx.x;
    if (i >= N_ENT * DIM) return;
    float dg = deg[i >> 7];
    dg = dg < 1.0f ? 1.0f : dg;
    float v = nbr[i] / dg + selfp[i];
    nbr[i] = v > 0.0f ? v : 0.0f;
}

// ---------------------------------------------------------------------------
// Entity GRU combine: h_out = (1-z)*n + z*h  (gates from precomputed xi, hh).
// ---------------------------------------------------------------------------
__global__ __launch_bounds__(256) void ent_gru_k(const float* __restrict__ h,
                                                 const float* __restrict__ xi,
                                                 const float* __restrict__ hh,
                                                 const float* __restrict__ bb,
                                                 float* __restrict__ h_out) {
    int i = blockIdx.x * 256 + threadIdx.x;
    if (i >= N_ENT * DIM) return;
    int row = i >> 7, d = i & 127;
    const float* xir = xi + (size_t)row * 3 * DIM;
    const float* hhr = hh + (size_t)row * 3 * DIM;
    float rg = 1.0f / (1.0f + __expf(-(xir[d] + hhr[d] + bb[d])));
    float zg = 1.0f / (1.0f + __expf(-(xir[DIM + d] + hhr[DIM + d] + bb[DIM + d])));
    float ng = tanhf(xir[2*DIM + d] + rg * hhr[2*DIM + d] + bb[2*DIM + d]);
    h_out[i] = (1.0f - zg) * ng + zg * h[i];
}

// ---------------------------------------------------------------------------
extern "C" void kernel_launch(void* const* d_in, const int* in_sizes, int n_in,
                              void* d_out, int out_size, void* d_ws, size_t ws_size,
                              hipStream_t stream) {
    const float* entity_emb   = (const float*)d_in[0];
    const float* relation_emb = (const float*)d_in[1];
    const float* basis        = (const float*)d_in[2];
    const float* comp         = (const float*)d_in[3];
    const float* self_w       = (const float*)d_in[4];
    const float* proj_w       = (const float*)d_in[5];
    const float* proj_b       = (const float*)d_in[6];
    const float* ent_wi       = (const float*)d_in[7];
    const float* ent_wh       = (const float*)d_in[8];
    const float* ent_b        = (const float*)d_in[9];
    const float* rel_wi       = (const float*)d_in[10];
    const float* rel_wh       = (const float*)d_in[11];
    const float* rel_b        = (const float*)d_in[12];
    const int*   edge_index   = (const int*)d_in[13];
    const int*   edge_type    = (const int*)d_in[14];
    const float* edge_mask    = (const float*)d_in[15];

    // ---- workspace layout (~134 MB) ----
    char* ws = (char*)d_ws;
    size_t off = 0;
    auto alloc = [&](size_t bytes) -> void* {
        void* p = ws + off;
        off += (bytes + 255) & ~(size_t)255;
        return p;
    };
    const size_t ND = (size_t)N_ENT * DIM;
    float* hA      = (float*)alloc(ND * 4);
    float* hB      = (float*)alloc(ND * 4);
    float* nbrA    = (float*)alloc(ND * 4);
    float* nbrB    = (float*)alloc(ND * 4);
    float* selfp   = (float*)alloc(ND * 4);
    float* agg     = (float*)alloc((size_t)N_ENT * DIM * NB * 4);   // 82 MB
    float* deg     = (float*)alloc((size_t)N_ENT * 4);
    float* rA      = (float*)alloc((size_t)N_REL2 * DIM * 4);
    float* rB      = (float*)alloc((size_t)N_REL2 * DIM * 4);
    float* rel_sum = (float*)alloc((size_t)N_REL2 * DIM * 4);
    float* rel_cnt = (float*)alloc((size_t)N_REL2 * 4);
    float* Wb      = (float*)alloc((size_t)NL * DIM * NB * DIM * 4);
    // GRU gate temps alias the agg buffer (never live simultaneously):
    float* xi = agg;                                   // N x 384
    float* hh = agg + (size_t)N_ENT * 3 * DIM;         // N x 384

    // ---- one-time setup (deterministic, re-done every call) ----
    hipMemcpyAsync(hA, entity_emb,   ND * 4,                    hipMemcpyDeviceToDevice, stream);
    hipMemcpyAsync(rA, relation_emb, (size_t)N_REL2 * DIM * 4,  hipMemcpyDeviceToDevice, stream);
    repack_basis_k<<<(NL * DIM * NB * DIM + 255) / 256, 256, 0, stream>>>(basis, Wb);

    float* hCur = hA; float* hNxt = hB;
    float* rCur = rA; float* rNxt = rB;

    const int MT  = N_ENT / 16;               // 1250 row tiles
    const int MB  = (MT + 7) / 8;             // 157 row blocks

    for (int t = 0; t < T_SNAP; ++t) {
        const int*   src = edge_index + (size_t)(t * 2 + 0) * NE;
        const int*   dst = edge_index + (size_t)(t * 2 + 1) * NE;
        const int*   et  = edge_type + (size_t)t * NE;
        const float* em  = edge_mask + (size_t)t * NE;

        // relation context + relation GRU (uses pre-update h)
        hipMemsetAsync(rel_sum, 0, (size_t)N_REL2 * DIM * 4, stream);
        hipMemsetAsync(rel_cnt, 0, (size_t)N_REL2 * 4, stream);
        hipMemsetAsync(deg, 0, (size_t)N_ENT * 4, stream);
        rel_scatter_k<<<(NE + 7) / 8, 256, 0, stream>>>(hCur, src, et, em, rel_sum, rel_cnt);
        rel_gru_k<<<N_REL2, 128, 0, stream>>>(rel_sum, rel_cnt, proj_w, proj_b,
                                              rCur, rel_wi, rel_wh, rel_b, rNxt);

        // R-GCN layers
        const float* xin = hCur;
        for (int l = 0; l < NL; ++l) {
            float* xout = (l == 0) ? nbrA : nbrB;
            hipMemsetAsync(agg, 0, (size_t)N_ENT * DIM * NB * 4, stream);
            edge_scatter_k<<<(NE + 1) / 2, 256, 0, stream>>>(
                xin, src, dst, et, em, comp + (size_t)l * N_REL2 * NB, agg, deg,
                (l == 0) ? 1 : 0);
            // nbr = agg (N x 1024) @ Wb_l (1024 x 128)
            gemm_wmma_k<<<MB * (DIM / 64), 256, 0, stream>>>(
                agg, Wb + (size_t)l * DIM * NB * DIM, xout, N_ENT, DIM * NB, DIM);
            // selfp = xin (N x 128) @ self_w_l (128 x 128)
            gemm_wmma_k<<<MB * (DIM / 64), 256, 0, stream>>>(
                xin, self_w + (size_t)l * DIM * DIM, selfp, N_ENT, DIM, DIM);
            combine_layer_k<<<(N_ENT * DIM + 255) / 256, 256, 0, stream>>>(xout, selfp, deg);
            xin = xout;
        }

        // entity GRU: xi = x @ ent_wi, hh = h @ ent_wh (both N x 384)
        gemm_wmma_k<<<MB * ((3 * DIM) / 64), 256, 0, stream>>>(xin,  ent_wi, xi, N_ENT, DIM, 3 * DIM);
        gemm_wmma_k<<<MB * ((3 * DIM) / 64), 256, 0, stream>>>(hCur, ent_wh, hh, N_ENT, DIM, 3 * DIM);
        ent_gru_k<<<(N_ENT * DIM + 255) / 256, 256, 0, stream>>>(hCur, xi, hh, ent_b, hNxt);

        { float* tmp = hCur; hCur = hNxt; hNxt = tmp; }
        { float* tmp = rCur; rCur = rNxt; rNxt = tmp; }
    }

    hipMemcpyAsync(d_out, hCur, ND * 4, hipMemcpyDeviceToDevice, stream);
}